// RelationModule_21706764714077
// MI455X (gfx1250) — compile-verified
//
#include <hip/hip_runtime.h>
#include <math.h>

typedef float v2f __attribute__((ext_vector_type(2)));
typedef float v8f __attribute__((ext_vector_type(8)));

#define BATCH   128
#define FEAT    1024
#define TWOFEAT 2048
#define CLS     500
#define CLS_PAD 512
#define HID     512
#define KCH     128   // h-chunk staged in LDS for the fused stage

// ---------------------------------------------------------------------------
// GEMM: out[m, n] = sum_k A[m, k] * W1[n, koff + k]
//   A   : Mval x FEAT, row-major. Rows are clamped (not zeroed) to stay in
//         bounds; tiles covering rows >= Mval produce garbage that is never
//         consumed downstream (output store for c >= CLS is guarded).
//   W1  : HID x 2048,  row-major
//   out : (gridDim.y*16) x HID, row-major workspace
// One wave per 16(M) x 64(N) output tile: 4 fp32 WMMA accumulators share one
// A fragment per K-step -> branch-free inner loop, 5 b64 loads : 4 WMMAs.
// ---------------------------------------------------------------------------
__global__ __launch_bounds__(32)
void fc1_gemm_wmma(const float* __restrict__ A, const float* __restrict__ W1,
                   float* __restrict__ out, int Mval, int koff)
{
    const int lane = threadIdx.x;        // 0..31
    const int lo   = lane & 15;
    const int hi   = lane >> 4;          // 0 or 1 (selects K-pair 0,1 vs 2,3)
    const int row0 = blockIdx.y * 16;
    const int col0 = blockIdx.x * 64;

    // Clamp instead of predicate: keeps EXEC uniform through the WMMA loop.
    const int m  = row0 + lo;
    const int mc = (m < Mval) ? m : (Mval - 1);

    const float* __restrict__ arow = A  + (size_t)mc * FEAT;
    const float* __restrict__ brow = W1 + (size_t)(col0 + lo) * TWOFEAT + koff;

    v8f acc0 = {}, acc1 = {}, acc2 = {}, acc3 = {};
    #pragma unroll 2
    for (int k0 = 0; k0 < FEAT; k0 += 4) {
        const int kb = k0 + hi * 2;      // lanes 16-31 take K=2,3 of the step
        const v2f a  = *(const v2f*)(arow + kb);
        const v2f b0 = *(const v2f*)(brow + kb);                 // col tile 0
        const v2f b1 = *(const v2f*)(brow + 16 * TWOFEAT + kb);  // col tile 1
        const v2f b2 = *(const v2f*)(brow + 32 * TWOFEAT + kb);  // col tile 2
        const v2f b3 = *(const v2f*)(brow + 48 * TWOFEAT + kb);  // col tile 3
        // 8 args: (neg_a, A, neg_b, B, c_mod, C, reuse_a, reuse_b)
        acc0 = __builtin_amdgcn_wmma_f32_16x16x4_f32(false, a, false, b0, (short)0, acc0, false, false);
        acc1 = __builtin_amdgcn_wmma_f32_16x16x4_f32(false, a, false, b1, (short)0, acc1, false, false);
        acc2 = __builtin_amdgcn_wmma_f32_16x16x4_f32(false, a, false, b2, (short)0, acc2, false, false);
        acc3 = __builtin_amdgcn_wmma_f32_16x16x4_f32(false, a, false, b3, (short)0, acc3, false, false);
    }

    // D layout: VGPR v -> row (row0 + v + hi*8), col (col_tile_base + lo)
    float* __restrict__ crow = out + (size_t)(row0 + hi * 8) * HID + (col0 + lo);
    #pragma unroll
    for (int v = 0; v < 8; ++v) {
        crow[(size_t)v * HID +  0] = acc0[v];
        crow[(size_t)v * HID + 16] = acc1[v];
        crow[(size_t)v * HID + 32] = acc2[v];
        crow[(size_t)v * HID + 48] = acc3[v];
    }
}

// ---------------------------------------------------------------------------
// Fused: out[b,c] = sigmoid(b2 + sum_h relu(imgp[b,h] + attrp[c,h] + b1[h]) * W2[h])
// Block = 256 threads = one 16(b) x 16(c) output tile; h staged in LDS chunks.
// attrp rows 500..511 hold garbage (clamped GEMM rows); they are loaded into
// LDS but only consumed by threads whose c >= CLS output store is guarded off.
// ---------------------------------------------------------------------------
__global__ __launch_bounds__(256)
void relation_fuse_kernel(const float* __restrict__ imgp,
                          const float* __restrict__ attrp,
                          const float* __restrict__ b1,
                          const float* __restrict__ W2,
                          const float* __restrict__ b2,
                          float* __restrict__ out)
{
    __shared__ float s_img[16][KCH + 1];   // +1 pad: row stride 129 -> no bank conflicts
    __shared__ float s_att[16][KCH + 1];
    __shared__ float s_w2[KCH];
    __shared__ float s_b1[KCH];

    const int tid = threadIdx.x;
    const int tb  = tid & 15;              // which b within tile
    const int tc  = tid >> 4;              // which c within tile
    const int b0  = blockIdx.y * 16;
    const int c0  = blockIdx.x * 16;

    float acc = 0.0f;

    for (int h0 = 0; h0 < HID; h0 += KCH) {
        // Cooperative, coalesced tile loads: 16*KCH floats each, 8 per thread.
        #pragma unroll
        for (int i = tid; i < 16 * KCH; i += 256) {
            const int r = i >> 7;          // i / KCH   (KCH == 128)
            const int h = i & (KCH - 1);   // i % KCH
            s_img[r][h] = imgp [(size_t)(b0 + r) * HID + h0 + h];
            s_att[r][h] = attrp[(size_t)(c0 + r) * HID + h0 + h];
        }
        if (tid < KCH) {
            s_w2[tid] = W2[h0 + tid];
            s_b1[tid] = b1[h0 + tid];
        }
        __syncthreads();

        #pragma unroll 8
        for (int h = 0; h < KCH; ++h) {
            float t = s_img[tb][h] + s_att[tc][h] + s_b1[h];
            t = fmaxf(t, 0.0f);
            acc = fmaf(t, s_w2[h], acc);
        }
        __syncthreads();
    }

    const int b = b0 + tb;
    const int c = c0 + tc;
    if (c < CLS) {
        const float x = acc + b2[0];
        out[(size_t)b * CLS + c] = 1.0f / (1.0f + __expf(-x));
    }
}

// ---------------------------------------------------------------------------
extern "C" void kernel_launch(void* const* d_in, const int* in_sizes, int n_in,
                              void* d_out, int out_size, void* d_ws, size_t ws_size,
                              hipStream_t stream)
{
    (void)in_sizes; (void)n_in; (void)out_size; (void)ws_size;

    const float* img  = (const float*)d_in[0];   // (128, 1024)
    const float* attr = (const float*)d_in[1];   // (500, 1024)
    const float* W1   = (const float*)d_in[2];   // (512, 2048)
    const float* b1   = (const float*)d_in[3];   // (512,)
    const float* W2   = (const float*)d_in[4];   // (1, 512)
    const float* b2   = (const float*)d_in[5];   // (1,)
    float* out = (float*)d_out;                  // (128, 500)

    float* imgp  = (float*)d_ws;                 // (128, 512)
    float* attrp = imgp + (size_t)BATCH * HID;   // (512, 512), rows >= 500 unused garbage

    // GEMM1: img_part = image_feats @ W1[:, :F].T    -> grid 8 x 8 waves
    dim3 g1(HID / 64, BATCH / 16);
    fc1_gemm_wmma<<<g1, 32, 0, stream>>>(img, W1, imgp, BATCH, 0);

    // GEMM2: attr_part = attributes @ W1[:, F:].T    -> grid 8 x 32 waves (padded M)
    dim3 g2(HID / 64, CLS_PAD / 16);
    fc1_gemm_wmma<<<g2, 32, 0, stream>>>(attr, W1, attrp, CLS, FEAT);

    // Fused broadcast-add + ReLU + W2-dot + sigmoid  -> grid 32 x 8 blocks of 256
    dim3 g3((CLS + 15) / 16, BATCH / 16);
    relation_fuse_kernel<<<g3, 256, 0, stream>>>(imgp, attrp, b1, W2, b2, out);
}